// SparsePrompter_uncertainty_360777253059
// MI455X (gfx1250) — compile-verified
//
#include <hip/hip_runtime.h>
#include <stdint.h>

// ---------------------------------------------------------------------------
// SparsePrompter: out[b,c,h,w] = x[b,c,h,w] + (rank(h,w) < PNUM ? prompts[rank, c] : 0)
// where rank(h,w) = descending rank of unc[h,w] (stable tie-break by index),
// exactly reproducing top_k ordering: prompt p lands at the pixel of rank p.
// ---------------------------------------------------------------------------

typedef __attribute__((ext_vector_type(4))) float        f32x4;
typedef __attribute__((ext_vector_type(4))) unsigned int u32x4;
// exact pointee type expected by __builtin_amdgcn_global_load_async_to_lds_b128
typedef __attribute__((__vector_size__(4 * sizeof(int)))) int v4i;

#define AS1 __attribute__((address_space(1)))
#define AS3 __attribute__((address_space(3)))

constexpr int H     = 135;
constexpr int W     = 240;
constexpr int HW    = H * W;            // 32400
constexpr int PNUM  = 12960;            // int(540*960*0.025)
constexpr int CCH   = 320;
constexpr int BATCH = 4;
constexpr int NELEM = BATCH * CCH * HW; // 41,472,000
constexpr int NV4   = NELEM / 4;        // 10,368,000 (HW % 4 == 0)
constexpr int CHUNK = 2048;             // comparison chunk staged in LDS
constexpr int NGRP  = (HW + CHUNK - 1) / CHUNK; // 16

// CDNA5 async global->LDS path (ASYNCcnt-tracked), guarded so we always compile.
#if __has_builtin(__builtin_amdgcn_global_load_async_to_lds_b128) && \
    __has_builtin(__builtin_amdgcn_s_wait_asynccnt)
#define HAVE_ASYNC_LDS 1
#endif

__device__ __forceinline__ void stage16_to_lds(const float* g, float* l) {
#ifdef HAVE_ASYNC_LDS
  // global_load_async_to_lds_b128: per-lane 16B DMA, tracked with ASYNCcnt
  __builtin_amdgcn_global_load_async_to_lds_b128(
      (AS1 v4i*)(uintptr_t)g,
      (AS3 v4i*)(uint32_t)(uintptr_t)l,  // low 32 bits of flat LDS addr == LDS offset
      0, 0);
#else
  *(f32x4*)l = *(const f32x4*)g;
#endif
}

__device__ __forceinline__ void async_stage_wait() {
#ifdef HAVE_ASYNC_LDS
  __builtin_amdgcn_s_wait_asynccnt(0);
#endif
}

// ---------------------------------------------------------------------------
// Kernel 1: partial rank counts. grid = (ceil(HW/256), NGRP).
// Block (bx, by) compares pixels [bx*256, bx*256+256) against unc chunk
// [by*CHUNK, by*CHUNK+len) staged in LDS; atomically accumulates into rank[].
// rank(i) = #{j: u[j] > u[i]}  +  #{j < i: u[j] == u[i]}
// ---------------------------------------------------------------------------
__global__ __launch_bounds__(256) void rank_count_kernel(
    const float* __restrict__ unc, unsigned int* __restrict__ rank) {
  __shared__ __align__(16) float tile[CHUNK];

  const int i    = blockIdx.x * 256 + threadIdx.x;
  const int base = blockIdx.y * CHUNK;
  const int len  = min(CHUNK, HW - base);
  const float v  = (i < HW) ? unc[i] : 0.0f;

  if (len == CHUNK) {
    // two b128 async DMAs per thread cover 2048 floats (16B-aligned both sides)
    const int t4 = threadIdx.x * 4;
    stage16_to_lds(unc + base + t4,        &tile[t4]);
    stage16_to_lds(unc + base + 1024 + t4, &tile[1024 + t4]);
    async_stage_wait();
  } else {
    for (int j = threadIdx.x; j < len; j += 256) tile[j] = unc[base + j];
  }
  __syncthreads();

  unsigned int cnt = 0;
#pragma unroll 4
  for (int j = 0; j < len; ++j) {
    const float u = tile[j];          // LDS broadcast read (conflict-free)
    const int  gj = base + j;
    cnt += (u > v) || ((u == v) && (gj < i));
  }

  if (i < HW) atomicAdd(&rank[i], cnt);
}

// ---------------------------------------------------------------------------
// Kernel 2: streaming copy + dense gather-add. One thread per float4.
// NT hints on the 332MB x/out streams keep L2 resident for prompts (16.6MB)
// and the rank map (130KB).
// ---------------------------------------------------------------------------
__global__ __launch_bounds__(256) void apply_prompts_kernel(
    const f32x4* __restrict__ x4,
    const float* __restrict__ prompts,       // [PNUM, CCH] row-major
    const unsigned int* __restrict__ rank,   // [HW]
    f32x4* __restrict__ out4) {
  const int tid = blockIdx.x * 256 + threadIdx.x;
  if (tid >= NV4) return;

  const int e0    = tid << 2;          // flat element index (< 2^31)
  const int plane = e0 / HW;           // b*CCH + c
  const int hw0   = e0 - plane * HW;   // 4-aligned pixel index
  const int c     = plane % CCH;

  f32x4 xv = __builtin_nontemporal_load(x4 + tid);
  const u32x4 r4 = *(const u32x4*)(rank + hw0);   // hot in cache, reused 1280x

  if (r4.x < (unsigned)PNUM) xv.x += prompts[r4.x * CCH + c];
  if (r4.y < (unsigned)PNUM) xv.y += prompts[r4.y * CCH + c];
  if (r4.z < (unsigned)PNUM) xv.z += prompts[r4.z * CCH + c];
  if (r4.w < (unsigned)PNUM) xv.w += prompts[r4.w * CCH + c];

  __builtin_nontemporal_store(xv, out4 + tid);
}

// ---------------------------------------------------------------------------
extern "C" void kernel_launch(void* const* d_in, const int* in_sizes, int n_in,
                              void* d_out, int out_size, void* d_ws, size_t ws_size,
                              hipStream_t stream) {
  const float* x       = (const float*)d_in[0];  // 41,472,000 fp32
  const float* unc     = (const float*)d_in[1];  // 32,400 fp32
  const float* prompts = (const float*)d_in[2];  // 12,960*320 fp32

  unsigned int* rank = (unsigned int*)d_ws;      // HW u32 accumulators

  (void)hipMemsetAsync(rank, 0, HW * sizeof(unsigned int), stream);

  dim3 g1((HW + 255) / 256, NGRP);               // 127 x 16 blocks
  rank_count_kernel<<<g1, 256, 0, stream>>>(unc, rank);

  apply_prompts_kernel<<<(NV4 + 255) / 256, 256, 0, stream>>>(
      (const f32x4*)x, prompts, rank, (f32x4*)d_out);
}